// NeuroPlasticLite_86569360818354
// MI455X (gfx1250) — compile-verified
//
#include <hip/hip_runtime.h>
#include <math.h>

#define NN    256
#define DD    32
#define KF    16
#define KN    50
#define BB    512
#define UINX  128
#define STEPS 20
#define GAMMA 0.1f
#define DTs   0.05f

typedef float v2f __attribute__((ext_vector_type(2)));
typedef float v8f __attribute__((ext_vector_type(8)));

// ---------------------------------------------------------------------------
// Kernel 1: top-k(50) of cosine similarity rows. One block per row n.
// Matches jax.lax.top_k ordering: descending value, ties -> lower index.
// ---------------------------------------------------------------------------
__global__ void __launch_bounds__(NN)
topk_kernel(const float* __restrict__ feat,
            float* __restrict__ vals, int* __restrict__ idxo) {
    const int n = blockIdx.x;
    const int t = threadIdx.x;   // column m
    __shared__ float sv[NN];
    __shared__ float rv[NN];
    __shared__ int   ri[NN];

    float dot = 0.f, nm2 = 0.f, nn2 = 0.f;
    for (int j = 0; j < KF; ++j) {
        float fm = feat[t * KF + j];
        float fn = feat[n * KF + j];
        dot += fm * fn; nm2 += fm * fm; nn2 += fn * fn;
    }
    sv[t] = dot / (sqrtf(nm2) * sqrtf(nn2));
    __syncthreads();

    for (int k = 0; k < KN; ++k) {
        rv[t] = sv[t]; ri[t] = t;
        __syncthreads();
        for (int off = NN / 2; off > 0; off >>= 1) {
            if (t < off) {
                float v1 = rv[t + off]; int i1 = ri[t + off];
                if (v1 > rv[t] || (v1 == rv[t] && i1 < ri[t])) { rv[t] = v1; ri[t] = i1; }
            }
            __syncthreads();
        }
        if (t == 0) {
            vals[n * KN + k] = rv[0];
            idxo[n * KN + k] = ri[0];
            sv[ri[0]] = -3.4e38f;
        }
        __syncthreads();
    }
}

// ---------------------------------------------------------------------------
// Kernel 2: u_proj = u @ w_in^T + b_in via V_WMMA_F32_16X16X4_F32.
// M = B*N = 131072 rows, K = 128, N = 32 (two 16-col tiles per wave).
// A lane layout (16x4 f32): lane<16 holds K0,K1; lane>=16 holds K2,K3.
// B lane layout (4x16 f32): lane<16 holds K0,K1 rows at col=lane; lane>=16 K2,K3.
// ---------------------------------------------------------------------------
__global__ void __launch_bounds__(256)
uproj_kernel(const float* __restrict__ u,
             const float* __restrict__ w_in,
             const float* __restrict__ b_in,
             float* __restrict__ uproj) {
    const int lane = threadIdx.x & 31;
    const int wid  = blockIdx.x * (blockDim.x >> 5) + (threadIdx.x >> 5);
    const int m0   = wid * 16;
    const int half = lane >> 4;
    const int l15  = lane & 15;

    v8f acc0 = {}; v8f acc1 = {};
    const float* urow = u    + (size_t)(m0 + l15) * UINX + 2 * half;
    const float* w0   = w_in + (size_t)l15        * UINX + 2 * half;
    const float* w1   = w_in + (size_t)(16 + l15) * UINX + 2 * half;

    for (int k0 = 0; k0 < UINX; k0 += 4) {
        v2f va  = *(const v2f*)(urow + k0);
        v2f vb0 = *(const v2f*)(w0 + k0);
        v2f vb1 = *(const v2f*)(w1 + k0);
        acc0 = __builtin_amdgcn_wmma_f32_16x16x4_f32(false, va, false, vb0,
                                                     (short)0, acc0, false, false);
        acc1 = __builtin_amdgcn_wmma_f32_16x16x4_f32(false, va, false, vb1,
                                                     (short)0, acc1, false, false);
    }
    const float bin0 = b_in[l15], bin1 = b_in[16 + l15];
    for (int i = 0; i < 8; ++i) {
        int row = m0 + i + 8 * half;
        uproj[(size_t)row * DD + l15]      = acc0[i] + bin0;
        uproj[(size_t)row * DD + 16 + l15] = acc1[i] + bin1;
    }
}

// ---------------------------------------------------------------------------
// Kernel 3: 20 dynamics steps. One block per batch element; N=256 neurons live
// in LDS so the a[:, idx] gather is intra-block. a_bar is dead code (output is
// only x) and dropped. Sigma-MLP second layer runs on WMMA f32 16x16x4.
// ---------------------------------------------------------------------------
__global__ void __launch_bounds__(NN)
step_kernel(const float* __restrict__ bias,
            const float* __restrict__ sig_w1,
            const float* __restrict__ sig_b1,
            const float* __restrict__ sig_w2,
            const float* __restrict__ sig_b2,
            const float* __restrict__ vals,
            const int*   __restrict__ idx,
            const float* uproj,   // aliases d_out, no restrict
            float*       out) {
    const int b = blockIdx.x;
    const int t = threadIdx.x;           // neuron id in phases 1-3

    __shared__ float xs[NN][DD + 1];     // padded: stride 33 -> conflict-free
    __shared__ float hs[NN][KF + 1];     // padded: stride 17
    __shared__ float as_[NN];

    const int lane = t & 31;
    const int wv   = t >> 5;             // wave 0..7
    const int half = lane >> 4;
    const int l15  = lane & 15;

    // ---- step-invariant preloads ----
    // sig_w2 B-tiles: vb[ntile][k-step]
    v2f vb[2][4];
    for (int kk = 0; kk < 4; ++kk) {
        int kb = kk * 4 + 2 * half;
        vb[0][kk] = *(const v2f*)(sig_w2 + l15        * KF + kb);
        vb[1][kk] = *(const v2f*)(sig_w2 + (16 + l15) * KF + kb);
    }
    // cu = bias + u_proj + sig_b2 at the positions this lane updates
    float cu[2][2][8];
    for (int mt = 0; mt < 2; ++mt) {
        int m0 = (wv * 2 + mt) * 16;
        for (int i = 0; i < 8; ++i) {
            int row = m0 + i + 8 * half;
            cu[mt][0][i] = bias[row * DD + l15] + sig_b2[l15]
                         + uproj[((size_t)b * NN + row) * DD + l15];
            cu[mt][1][i] = bias[row * DD + 16 + l15] + sig_b2[16 + l15]
                         + uproj[((size_t)b * NN + row) * DD + 16 + l15];
        }
    }
    float w1r[KF], b1r[KF];
    for (int j = 0; j < KF; ++j) { w1r[j] = sig_w1[j]; b1r[j] = sig_b1[j]; }

    // x0 = 0
    for (int d = 0; d < DD; ++d) xs[t][d] = 0.f;
    __syncthreads();

    const float* vrow = vals + t * KN;
    const int*   irow = idx  + t * KN;

    for (int s = 0; s < STEPS; ++s) {
        // phase 1: firing rates a = tanh(||x||)
        float s2 = 0.f;
        for (int d = 0; d < DD; ++d) { float v = xs[t][d]; s2 += v * v; }
        as_[t] = tanhf(sqrtf(s2 + 1e-12f));
        __syncthreads();

        // phase 2+3: sparse neighbor aggregate + sigma layer-1 (exact GELU)
        float syn = 0.f;
        for (int k = 0; k < KN; ++k) syn += as_[irow[k]] * vrow[k];
        for (int j = 0; j < KF; ++j) {
            float z = syn * w1r[j] + b1r[j];
            hs[t][j] = 0.5f * z * (1.0f + erff(z * 0.70710678118654752f));
        }
        __syncthreads();

        // phase 4: h @ sig_w2^T via WMMA, then Euler update of x in LDS
        for (int mt = 0; mt < 2; ++mt) {
            int m0 = (wv * 2 + mt) * 16;
            v8f acc0 = {}; v8f acc1 = {};
            for (int kk = 0; kk < 4; ++kk) {
                int kb = kk * 4 + 2 * half;
                v2f va;
                va.x = hs[m0 + l15][kb];
                va.y = hs[m0 + l15][kb + 1];
                acc0 = __builtin_amdgcn_wmma_f32_16x16x4_f32(false, va, false, vb[0][kk],
                                                             (short)0, acc0, false, false);
                acc1 = __builtin_amdgcn_wmma_f32_16x16x4_f32(false, va, false, vb[1][kk],
                                                             (short)0, acc1, false, false);
            }
            for (int i = 0; i < 8; ++i) {
                int row = m0 + i + 8 * half;
                float x0 = xs[row][l15];
                xs[row][l15]      = x0 + DTs * (-GAMMA * x0 + acc0[i] + cu[mt][0][i]);
                float x1 = xs[row][16 + l15];
                xs[row][16 + l15] = x1 + DTs * (-GAMMA * x1 + acc1[i] + cu[mt][1][i]);
            }
        }
        __syncthreads();
    }

    // write final state (overwrites the u_proj staging for this b — already consumed)
    for (int d = 0; d < DD; ++d)
        out[((size_t)b * NN + t) * DD + d] = xs[t][d];
}

// ---------------------------------------------------------------------------
extern "C" void kernel_launch(void* const* d_in, const int* in_sizes, int n_in,
                              void* d_out, int out_size, void* d_ws, size_t ws_size,
                              hipStream_t stream) {
    const float* u    = (const float*)d_in[0];
    const float* feat = (const float*)d_in[1];
    const float* bias = (const float*)d_in[2];
    const float* w_in = (const float*)d_in[3];
    const float* b_in = (const float*)d_in[4];
    const float* sw1  = (const float*)d_in[5];
    const float* sb1  = (const float*)d_in[6];
    const float* sw2  = (const float*)d_in[7];
    const float* sb2  = (const float*)d_in[8];
    float* out = (float*)d_out;

    float* vals = (float*)d_ws;
    int*   idx  = (int*)((char*)d_ws + (size_t)NN * KN * sizeof(float));
    float* uproj = out;  // stage u_proj in d_out; fully consumed per-block before final x store

    topk_kernel<<<NN, NN, 0, stream>>>(feat, vals, idx);
    uproj_kernel<<<(BB * NN) / (16 * 8), 256, 0, stream>>>(u, w_in, b_in, uproj);
    step_kernel<<<BB, NN, 0, stream>>>(bias, sw1, sb1, sw2, sb2, vals, idx, uproj, out);
}